// messagePassing_90615220011123
// MI455X (gfx1250) — compile-verified
//
#include <hip/hip_runtime.h>

#define D_FEAT 64
#define EDGES_PER_ITER 4

// ---------------------------------------------------------------------------
// Kernel 1: zero the output accumulator (harness poisons d_out with 0xAA and
// we accumulate with atomics, so every launch must start from zeros).
// float4 stores -> global_store_b128 (verified in round-1 asm).
// ---------------------------------------------------------------------------
__global__ __launch_bounds__(256) void mp_zero_kernel(float4* __restrict__ out,
                                                      int n4) {
    int i = blockIdx.x * blockDim.x + threadIdx.x;
    if (i < n4) {
        out[i] = make_float4(0.0f, 0.0f, 0.0f, 0.0f);
    }
}

// ---------------------------------------------------------------------------
// Kernel 2: gather + scatter-add, 4 edges per wave per iteration.
//   - src[e..e+3] / dst[e..e+3] are contiguous int64 -> scalar loads merge
//     into wide SMEM loads (s_load_b256), 1 scalar request per 4 edges.
//   - 4 independent global_load_b64 gathers issued back-to-back (MLP to
//     hide L2 latency), then 8 independent non-returning
//     global_atomic_add_f32 scatters (STOREcnt-tracked, wave never blocks
//     on them inside the loop).
// x (25.6 MB) and out (25.6 MB) both stay resident in the 192 MB L2, so the
// steady state is L2 atomic/load throughput bound, not HBM bound.
// ---------------------------------------------------------------------------
__global__ __launch_bounds__(256) void mp_scatter_add_kernel(
    const float* __restrict__ x,
    const long long* __restrict__ src,   // edge_index[0], int64
    const long long* __restrict__ dst,   // edge_index[1], int64
    float* __restrict__ out,
    int n_edges) {

    const int lane = threadIdx.x & 31;
    // Force the wave id into an SGPR so edge-index loads are provably
    // wave-uniform and lower to scalar (SMEM) loads.
    const int wave = __builtin_amdgcn_readfirstlane((int)(threadIdx.x >> 5));
    const int waves_per_block = (int)(blockDim.x >> 5);
    const int wave_stride = (int)gridDim.x * waves_per_block * EDGES_PER_ITER;

    const int feat_off = lane * 2;

    int e0 = ((int)blockIdx.x * waves_per_block + wave) * EDGES_PER_ITER;

    for (; e0 < n_edges; e0 += wave_stride) {
        // Prefetch next batch's index cachelines (uniform addresses).
        int en = e0 + wave_stride;
        if (en < n_edges) {
            __builtin_prefetch(&src[en], 0, 1);
            __builtin_prefetch(&dst[en], 0, 1);
        }

        if (e0 + (EDGES_PER_ITER - 1) < n_edges) {
            // 32 contiguous bytes of src indices + 32 of dst indices:
            // mergeable into wide scalar loads.
            const long long s0 = src[e0 + 0];
            const long long s1 = src[e0 + 1];
            const long long s2 = src[e0 + 2];
            const long long s3 = src[e0 + 3];
            const long long d0 = dst[e0 + 0];
            const long long d1 = dst[e0 + 1];
            const long long d2 = dst[e0 + 2];
            const long long d3 = dst[e0 + 3];

            // Four independent gathers: 256 B/row, one b64 per lane each.
            const float2 v0 = *reinterpret_cast<const float2*>(x + s0 * D_FEAT + feat_off);
            const float2 v1 = *reinterpret_cast<const float2*>(x + s1 * D_FEAT + feat_off);
            const float2 v2 = *reinterpret_cast<const float2*>(x + s2 * D_FEAT + feat_off);
            const float2 v3 = *reinterpret_cast<const float2*>(x + s3 * D_FEAT + feat_off);

            // Eight independent non-returning hw fp32 atomics.
            float* o0 = out + d0 * D_FEAT + feat_off;
            float* o1 = out + d1 * D_FEAT + feat_off;
            float* o2 = out + d2 * D_FEAT + feat_off;
            float* o3 = out + d3 * D_FEAT + feat_off;
            unsafeAtomicAdd(o0,     v0.x);
            unsafeAtomicAdd(o0 + 1, v0.y);
            unsafeAtomicAdd(o1,     v1.x);
            unsafeAtomicAdd(o1 + 1, v1.y);
            unsafeAtomicAdd(o2,     v2.x);
            unsafeAtomicAdd(o2 + 1, v2.y);
            unsafeAtomicAdd(o3,     v3.x);
            unsafeAtomicAdd(o3 + 1, v3.y);
        } else {
            // Tail: scalar per-edge loop.
            for (int e = e0; e < n_edges; ++e) {
                const long long s = src[e];
                const long long d = dst[e];
                const float2 v =
                    *reinterpret_cast<const float2*>(x + s * D_FEAT + feat_off);
                float* o = out + d * D_FEAT + feat_off;
                unsafeAtomicAdd(o,     v.x);
                unsafeAtomicAdd(o + 1, v.y);
            }
        }
    }
}

// ---------------------------------------------------------------------------
// Launch wrapper.
// ---------------------------------------------------------------------------
extern "C" void kernel_launch(void* const* d_in, const int* in_sizes, int n_in,
                              void* d_out, int out_size, void* d_ws, size_t ws_size,
                              hipStream_t stream) {
    const float* x = (const float*)d_in[0];                   // (N, 64) fp32
    const long long* edge_index = (const long long*)d_in[1];  // (2, E) int64

    const int n_edges = in_sizes[1] / 2;
    float* out = (float*)d_out;                               // (N, 64) fp32

    // Zero the accumulator every call (atomics accumulate on top).
    const int n4 = out_size / 4;                              // D=64 -> divisible
    const int zblocks = (n4 + 255) / 256;
    mp_zero_kernel<<<zblocks, 256, 0, stream>>>((float4*)d_out, n4);

    // 4096 blocks x 8 waves = 32768 waves, 4 edges per wave per iteration,
    // ~9-10 grid-stride sweeps over the 1.25M edges.
    const int sblocks = 4096;
    mp_scatter_add_kernel<<<sblocks, 256, 0, stream>>>(
        x, edge_index, edge_index + n_edges, out, n_edges);
}